// IntraVolume_Attention_34471407518351
// MI455X (gfx1250) — compile-verified
//
#include <hip/hip_runtime.h>
#include <math.h>

// CDNA5 / gfx1250: wave32, WMMA 16x16x4 FP32
typedef __attribute__((ext_vector_type(2))) float v2f;
typedef __attribute__((ext_vector_type(8))) float v8f;

#define NF 512    // n_filters (F)
#define DM 128    // d_model
#define NROWS 256 // B*V = 8*32
#define NT 512    // threads per block (16 waves -> 4 waves per SIMD32)
#define LOG2E 1.4426950408889634f

__global__ __launch_bounds__(NT) void iv_attn_wmma_kernel(
    const float* __restrict__ x,
    const float* __restrict__ wq,
    const float* __restrict__ wk,
    const float* __restrict__ wv,
    float* __restrict__ out)
{
    __shared__ float sx[NF];
    __shared__ float red[NT];
    __shared__ float redmin[NT];
    __shared__ float s_c, s_xmax, s_xmin, s_m;

    const int tid = threadIdx.x;
    const int row = blockIdx.x;
    const float* xr = x + (size_t)row * NF;

    // ---- c = dot(w_q, w_k) / sqrt(128): block reduction ----
    red[tid] = (tid < DM) ? wq[tid] * wk[tid] : 0.0f;
    __syncthreads();
    for (int s = NT / 2; s > 0; s >>= 1) {
        if (tid < s) red[tid] += red[tid + s];
        __syncthreads();
    }
    if (tid == 0) s_c = red[0] * 0.08838834764831845f; // 1/sqrt(128)
    __syncthreads();
    const float c = s_c;
    __syncthreads();

    // ---- stage x row in LDS; row min/max for exact softmax max-subtraction ----
    const float v = xr[tid];     // NT == NF: one element per thread
    sx[tid] = v;
    red[tid] = v;
    redmin[tid] = v;
    __syncthreads();
    for (int s = NT / 2; s > 0; s >>= 1) {
        if (tid < s) {
            red[tid]    = fmaxf(red[tid], red[tid + s]);
            redmin[tid] = fminf(redmin[tid], redmin[tid + s]);
        }
        __syncthreads();
    }
    if (tid == 0) { s_xmax = red[0]; s_xmin = redmin[0]; }
    __syncthreads();
    const float xmax = s_xmax, xmin = s_xmin;

    // ---- per-wave WMMA attention over f-tiles of 16 ----
    const int lane  = tid & 31;
    const int wave  = tid >> 5;         // 16 waves
    const int m16   = lane & 15;        // A: row M; B/D: column N
    const int khalf = (lane >> 4) << 1; // K offset: 0 (lanes 0-15) or 2 (lanes 16-31)
    const bool isN0 = (m16 == 0);
    const float cb  = (m16 == 1) ? 1.0f : 0.0f; // B column 1 = ones, cols >=2 zero

    float localSum = 0.0f;

    for (int tile = 0; tile < 2; ++tile) {          // 16 waves x 2 tiles x 16 = 512 f
        const int f    = (wave * 2 + tile) * 16 + m16;
        const float t  = c * sx[f];
        const float M  = (t > 0.0f) ? t * xmax : t * xmin; // exact max_g(t*x_g)
        // pre-scale by log2(e): exp(t*x - M) == exp2(tl*x - Ml)
        const float tl = t * LOG2E;
        const float Ml = M * LOG2E;

        v8f acc = {};
        #pragma unroll 8
        for (int g = 0; g < NF; g += 4) {
            const float xg0 = sx[g + khalf];
            const float xg1 = sx[g + khalf + 1];
            v2f a, b;
            a.x = __builtin_amdgcn_exp2f(__builtin_fmaf(tl, xg0, -Ml)); // A[m, khalf+0]
            a.y = __builtin_amdgcn_exp2f(__builtin_fmaf(tl, xg1, -Ml)); // A[m, khalf+1]
            b.x = isN0 ? xg0 : cb;                                      // B[khalf+0, n]
            b.y = isN0 ? xg1 : cb;                                      // B[khalf+1, n]
            // D = A(16x4 f32) * B(4x16 f32) + C, full fp32
            acc = __builtin_amdgcn_wmma_f32_16x16x4_f32(
                      false, a, false, b, (short)0, acc, false, false);
        }

        // D layout: lane N=0 holds num_f (col 0), lane N=1 holds den_f (col 1);
        // VGPR j -> f = tileBase + j + 8*(lane>>4)
        #pragma unroll
        for (int j = 0; j < 8; ++j) {
            float den = __shfl(acc[j], lane + 1, 32); // lanes 0/16 read lanes 1/17
            // den >= 1 always (max-subtracted softmax has an exp(0)=1 term):
            // fast reciprocal + one Newton-Raphson step, branch-free accumulate
            float r = __builtin_amdgcn_rcpf(den);
            r = r * __builtin_fmaf(-den, r, 2.0f);
            localSum += isN0 ? acc[j] * r : 0.0f;     // s_f = num_f / den_f
        }
    }

    // ---- mean over f, then out[row, :] = m * w_v ----
    red[tid] = localSum;
    __syncthreads();
    for (int s = NT / 2; s > 0; s >>= 1) {
        if (tid < s) red[tid] += red[tid + s];
        __syncthreads();
    }
    if (tid == 0) s_m = red[0] * (1.0f / (float)NF);
    __syncthreads();

    if (tid < DM)
        out[(size_t)row * DM + tid] = s_m * wv[tid];
}

extern "C" void kernel_launch(void* const* d_in, const int* in_sizes, int n_in,
                              void* d_out, int out_size, void* d_ws, size_t ws_size,
                              hipStream_t stream) {
    const float* x  = (const float*)d_in[0];
    const float* wq = (const float*)d_in[1];
    const float* wk = (const float*)d_in[2];
    const float* wv = (const float*)d_in[3];
    float* out = (float*)d_out;

    iv_attn_wmma_kernel<<<NROWS, NT, 0, stream>>>(x, wq, wk, wv, out);
}